// CombinedLoss_15960098472629
// MI455X (gfx1250) — compile-verified
//
#include <hip/hip_runtime.h>
#include <hip/hip_bf16.h>
#include <math.h>

typedef __attribute__((ext_vector_type(2))) float v2f;
typedef __attribute__((ext_vector_type(8))) float v8f;

// Problem constants (match reference setup_inputs)
#define BATCH 4
#define N_UP  8192
#define N_GT  8192
#define N_RAD 1024

#define QCHUNK 2048              // Q points staged in LDS per chunk (32 KB as float4)

// Accumulator layout in d_ws (floats):
// acc[0] = sum(dist1)  acc[1] = sum(sqrt(dist1))  acc[2] = sum(dist2)  acc[3] = conf SSE
__global__ void cl_init_kernel(float* __restrict__ acc) {
    if (threadIdx.x < 4) acc[threadIdx.x] = 0.0f;
}

// One wave = 16 rows of P (a strip). Q is staged through LDS in 2048-point
// chunks shared by all waves of the block, stored as float4 {x, y, z, |q|^2}
// (q^2 precomputed once in the cold staging path). Per 16x16 tile, one
// V_WMMA_F32_16X16X4_F32 computes  D = -2 p.q + |p|^2  (C = inline 0):
//   A row m = [-2px, -2py, -2pz, |p|^2] (K=4), B col n = [qx, qy, qz, 1]
// and |q_n|^2 is added afterwards, fused into the running min (pk_add + min3).
// mode 0: acc[0]+=sum(min), acc[1]+=sum(sqrt(min))       (P=pc_up, Q=pc2)
// mode 1: acc[2]+=sum(min)                               (P=pc2,  Q=pc_up)
// mode 2: acc[3]+=sum((conf - exp(-sqrt(min)))^2)        (P=pc3,  Q=pc2)
__global__ void cl_rowmin_kernel(const float* __restrict__ P, int NP,
                                 const float* __restrict__ Q, int NQ,
                                 const float* __restrict__ conf,
                                 float* __restrict__ acc, int mode)
{
    __shared__ float4 smem[QCHUNK];      // 32 KB: {x, y, z, |q|^2} per point

    const int tid           = threadIdx.x;
    const int lane          = tid & 31;
    const int wavesPerBlock = blockDim.x >> 5;
    const int wave          = blockIdx.x * wavesPerBlock + (tid >> 5);
    const int stripsPerB    = NP >> 4;
    const int b             = wave / stripsPerB;
    const int strip         = wave % stripsPerB;

    const float* Pr = P + ((size_t)b * NP + (size_t)strip * 16) * 3;
    const float* Qb = Q + (size_t)b * NQ * 3;

    const int half = lane >> 4;   // which K-half of the wave this lane is in
    const int idx  = lane & 15;   // row (for A) / column (for B) index

    // A-matrix 16x4 f32 layout: lanes 0-15 hold {K0,K1}, lanes 16-31 hold {K2,K3}
    const float px = Pr[idx * 3 + 0];
    const float py = Pr[idx * 3 + 1];
    const float pz = Pr[idx * 3 + 2];
    v2f a;
    a.x = (half == 0) ? (-2.0f * px) : (-2.0f * pz);
    a.y = (half == 0) ? (-2.0f * py) : (px * px + py * py + pz * pz);

    v8f rmin;
#pragma unroll
    for (int r = 0; r < 8; ++r) rmin[r] = 3.0e38f;

    const int nchunks = NQ / QCHUNK;
    for (int ch = 0; ch < nchunks; ++ch) {
        // Cold path: stage Q chunk into LDS, precomputing |q|^2 per point.
        {
            const float* src = Qb + (size_t)ch * QCHUNK * 3;
            for (int i = tid; i < QCHUNK; i += blockDim.x) {
                const float qx = src[i * 3 + 0];
                const float qy = src[i * 3 + 1];
                const float qz = src[i * 3 + 2];
                float4 v;
                v.x = qx; v.y = qy; v.z = qz;
                v.w = fmaf(qx, qx, fmaf(qy, qy, qz * qz));
                smem[i] = v;
            }
        }
        __syncthreads();

        const int ntiles = QCHUNK >> 4;                      // 128
        for (int t = 0; t < ntiles; ++t) {
            const float4 q = smem[t * 16 + idx];             // single ds_load_b128

            // B-matrix 4x16 f32: columns across lanes; K-halves mirror A's split
            v2f bm;
            bm.x = (half == 0) ? q.x : q.z;
            bm.y = (half == 0) ? q.y : 1.0f;

            v8f cz = {};  // inline-0 C operand
            v8f d = __builtin_amdgcn_wmma_f32_16x16x4_f32(
                /*neg_a=*/false, a, /*neg_b=*/false, bm,
                /*c_mod=*/(short)0, cz, /*reuse_a=*/false, /*reuse_b=*/false);

#pragma unroll
            for (int r = 0; r < 8; ++r)
                rmin[r] = fminf(rmin[r], d[r] + q.w);
        }
        __syncthreads();
    }

    // Min over the 16 columns: xor 1,2,4,8 stays inside each 16-lane half.
    // After this, lanes 0-15 hold row-mins for rows 0..7 (slot r = row r),
    // lanes 16-31 hold rows 8..15.
#pragma unroll
    for (int off = 8; off >= 1; off >>= 1) {
#pragma unroll
        for (int r = 0; r < 8; ++r)
            rmin[r] = fminf(rmin[r], __shfl_xor(rmin[r], off, 32));
    }
    // Deferred jnp.maximum(., 0) clamp: max commutes with the min-reduction.
#pragma unroll
    for (int r = 0; r < 8; ++r) rmin[r] = fmaxf(rmin[r], 0.0f);

    if (idx == 0) {  // lanes 0 and 16 each own 8 distinct rows
        if (mode == 0) {
            float s = 0.0f, ss = 0.0f;
#pragma unroll
            for (int r = 0; r < 8; ++r) { s += rmin[r]; ss += sqrtf(rmin[r]); }
            atomicAdd(&acc[0], s);
            atomicAdd(&acc[1], ss);
        } else if (mode == 1) {
            float s = 0.0f;
#pragma unroll
            for (int r = 0; r < 8; ++r) s += rmin[r];
            atomicAdd(&acc[2], s);
        } else {
            const float* cf = conf + (size_t)b * NP + (size_t)strip * 16 + half * 8;
            float sse = 0.0f;
#pragma unroll
            for (int r = 0; r < 8; ++r) {
                const float score = expf(-sqrtf(rmin[r]));
                const float e     = cf[r] - score;
                sse += e * e;
            }
            atomicAdd(&acc[3], sse);
        }
    }
}

__global__ void cl_finalize_kernel(const float* __restrict__ acc,
                                   float* __restrict__ out)
{
    const float mean_d1   = acc[0] / (float)(BATCH * N_UP);
    const float mean_sqrt = acc[1] / (float)(BATCH * N_UP);
    const float mean_d2   = acc[2] / (float)(BATCH * N_GT);
    const float conf_loss = acc[3] / (float)(BATCH * N_RAD);
    const float chamfer   = 0.5f * mean_d1 + 2.0f * mean_d2;
    out[0] = 0.5f * chamfer + 0.5f * conf_loss + mean_sqrt;  // ALPHA = 0.5
}

extern "C" void kernel_launch(void* const* d_in, const int* in_sizes, int n_in,
                              void* d_out, int out_size, void* d_ws, size_t ws_size,
                              hipStream_t stream) {
    (void)in_sizes; (void)n_in; (void)out_size; (void)ws_size;
    const float* pc_up   = (const float*)d_in[0];
    // d_in[1] = pc_seed (unused by the reference loss)
    const float* pc_conf = (const float*)d_in[2];
    const float* pc2     = (const float*)d_in[3];
    const float* pc3     = (const float*)d_in[4];
    float* acc = (float*)d_ws;
    float* out = (float*)d_out;

    cl_init_kernel<<<1, 32, 0, stream>>>(acc);

    const int wavesPerBlock = 4;                  // 128 threads = 4 SIMD32s of a WGP
    const int threads       = wavesPerBlock * 32;

    // dist1: up -> gt row mins (also feeds EMD surrogate)
    cl_rowmin_kernel<<<(BATCH * N_UP / 16) / wavesPerBlock, threads, 0, stream>>>(
        pc_up, N_UP, pc2, N_GT, nullptr, acc, 0);
    // dist2: gt -> up row mins (roles swapped; recompute beats atomic min traffic)
    cl_rowmin_kernel<<<(BATCH * N_GT / 16) / wavesPerBlock, threads, 0, stream>>>(
        pc2, N_GT, pc_up, N_UP, nullptr, acc, 1);
    // confidence loss: radar -> gt row mins
    cl_rowmin_kernel<<<(BATCH * N_RAD / 16) / wavesPerBlock, threads, 0, stream>>>(
        pc3, N_RAD, pc2, N_GT, pc_conf, acc, 2);

    cl_finalize_kernel<<<1, 1, 0, stream>>>(acc, out);
}